// L_spa_25409026523585
// MI455X (gfx1250) — compile-verified
//
#include <hip/hip_runtime.h>

typedef float v2f __attribute__((ext_vector_type(2)));
typedef float v4f __attribute__((ext_vector_type(4)));
typedef float v8f __attribute__((ext_vector_type(8)));

// Problem is fixed-shape per the reference: B=32, C=3, H=W=512 -> pooled 128x128.
#define BATCH 32
#define HW    512
#define POOL  128           // HW/4
#define GROUPS (BATCH * POOL * (POOL / 16))   // 16 outputs per wave -> 32768 groups

// ---------------------------------------------------------------------------
// Pass 1: p[b,yo,xo] = (1/48) * sum_{c<3, dy<4, dx<4} (org - enh)[b,c,4yo+dy,4xo+dx]
//
// One wave32 produces 16 consecutive xo outputs via 12 accumulating
// V_WMMA_F32_16X16X4_F32 ops:
//   A[m,k] = diff row-segment, m = output pixel (0..15), k = dx pair
//     (ISA layout: lanes 0-15 hold {K=0,K=1}, lanes 16-31 hold {K=2,K=3}
//      -> lane loads float2 at element offset 4*(lane&15) + 2*(lane>>4))
//   B = uniform 1/48 (uniform matrix => B's exact VGPR layout is irrelevant)
//   D[m,n] = p_m for every n; lane 0 holds p_0..p_7 in acc[0..7],
//   lane 16 holds p_8..p_15 (C/D layout: vgpr r, lanes16-31 = row 8+r).
// ---------------------------------------------------------------------------
__global__ __launch_bounds__(256) void spa_pool_diff_wmma(
    const float* __restrict__ org, const float* __restrict__ enh,
    float* __restrict__ p)
{
    const int lane  = threadIdx.x & 31;
    const int group = blockIdx.x * 8 + (threadIdx.x >> 5);   // 8 waves / block

    const int xg = group & 7;             // 16-pixel group along x
    const int yo = (group >> 3) & 127;
    const int b  = group >> 10;

    const int chanStride = HW * HW;
    const int rowBase = (b * 3) * chanStride + (4 * yo) * HW + 64 * xg;

#if __has_builtin(__builtin_amdgcn_wmma_f32_16x16x4_f32)
    const int laneOff = 4 * (lane & 15) + ((lane >> 4) << 1); // element offset in segment

    v8f acc = {0.f, 0.f, 0.f, 0.f, 0.f, 0.f, 0.f, 0.f};
    const v2f bmat = {1.0f / 48.0f, 1.0f / 48.0f};

    #pragma unroll
    for (int c = 0; c < 3; ++c) {
        #pragma unroll
        for (int dy = 0; dy < 4; ++dy) {
            const int off = rowBase + c * chanStride + dy * HW + laneOff;
            // Streaming inputs (201 MB, read once): non-temporal so the 2 MB
            // intermediate + L2 stay clean for pass 2.
            v2f o = __builtin_nontemporal_load((const v2f*)(org + off));
            v2f e = __builtin_nontemporal_load((const v2f*)(enh + off));
            v2f a; a.x = o.x - e.x; a.y = o.y - e.y;
            acc = __builtin_amdgcn_wmma_f32_16x16x4_f32(
                /*neg_a=*/false, a, /*neg_b=*/false, bmat,
                /*c_mod=*/(short)0, acc, /*reuse_a=*/false, /*reuse_b=*/false);
        }
    }

    float* outp = p + (b * POOL + yo) * POOL + 16 * xg;
    if (lane == 0) {
        v4f lo = {acc[0], acc[1], acc[2], acc[3]};
        v4f hi = {acc[4], acc[5], acc[6], acc[7]};
        *(v4f*)(outp + 0) = lo;
        *(v4f*)(outp + 4) = hi;
    } else if (lane == 16) {
        v4f lo = {acc[0], acc[1], acc[2], acc[3]};
        v4f hi = {acc[4], acc[5], acc[6], acc[7]};
        *(v4f*)(outp + 8)  = lo;
        *(v4f*)(outp + 12) = hi;
    }
#else
    // Scalar fallback (lanes 0-15 each reduce one output pixel directly).
    if (lane < 16) {
        float sum = 0.f;
        const int base = rowBase + 4 * lane;
        #pragma unroll
        for (int c = 0; c < 3; ++c)
            #pragma unroll
            for (int dy = 0; dy < 4; ++dy) {
                const float* po = org + base + c * chanStride + dy * HW;
                const float* pe = enh + base + c * chanStride + dy * HW;
                #pragma unroll
                for (int dx = 0; dx < 4; ++dx) sum += po[dx] - pe[dx];
            }
        p[(b * POOL + yo) * POOL + 16 * xg + lane] = sum * (1.0f / 48.0f);
    }
#endif
}

// ---------------------------------------------------------------------------
// Pass 2: E = (p-pL)^2 + (p-pR)^2 + (p-pU)^2 + (p-pD)^2, zero padding (SAME).
// p is 2 MB -> L2 resident; this pass is ~2% of total traffic.
// ---------------------------------------------------------------------------
__global__ __launch_bounds__(256) void spa_energy(
    const float* __restrict__ p, float* __restrict__ E)
{
    const int idx = blockIdx.x * 256 + threadIdx.x;   // 0 .. 524287
    const int j = idx & 127;
    const int i = (idx >> 7) & 127;
    const int b = idx >> 14;

    const float* pb = p + b * POOL * POOL;
    const float c = pb[i * POOL + j];
    const float l = (j > 0)          ? pb[i * POOL + j - 1] : 0.f;
    const float r = (j < POOL - 1)   ? pb[i * POOL + j + 1] : 0.f;
    const float u = (i > 0)          ? pb[(i - 1) * POOL + j] : 0.f;
    const float d = (i < POOL - 1)   ? pb[(i + 1) * POOL + j] : 0.f;

    const float dl = c - l, dr = c - r, du = c - u, dd = c - d;
    E[idx] = dl * dl + dr * dr + du * du + dd * dd;
}

extern "C" void kernel_launch(void* const* d_in, const int* in_sizes, int n_in,
                              void* d_out, int out_size, void* d_ws, size_t ws_size,
                              hipStream_t stream) {
    const float* org = (const float*)d_in[0];
    const float* enh = (const float*)d_in[1];
    float* p = (float*)d_ws;       // 32*128*128 floats = 2 MB scratch
    float* E = (float*)d_out;      // 524288 floats

    // Pass 1: 32768 wave-groups, 8 waves per 256-thread block -> 4096 blocks.
    spa_pool_diff_wmma<<<GROUPS / 8, 256, 0, stream>>>(org, enh, p);
    // Pass 2: one thread per output element.
    spa_energy<<<(BATCH * POOL * POOL) / 256, 256, 0, stream>>>(p, E);
}